// BSM_83236466196837
// MI455X (gfx1250) — compile-verified
//
#include <hip/hip_runtime.h>
#include <hip/hip_bf16.h>

typedef float v2f __attribute__((ext_vector_type(2)));
typedef float v4f __attribute__((ext_vector_type(4)));
typedef float v8f __attribute__((ext_vector_type(8)));

#define BSM_B    64
#define BSM_T    1024
#define BSM_C    768
#define BSM_CK   64
#define BSM_R    256
#define BSM_HALF 512
#define NEG_INF_F (-1e30f)

// ---------------------------------------------------------------------------
// Kernel 1: node_max / node_idx of scores = a @ b^T via V_WMMA_F32_16X16X4_F32
// One wave (32 threads) per (batch, 16-row s-tile). K=64 -> 16 chained WMMAs.
// B tile is loaded into a register array first so the 16 global_load_b64s
// issue as one clause (latency overlapped with the WMMA chain); unroll-2 on
// the tile loop lets the next tile's loads issue under the current chain.
// ---------------------------------------------------------------------------
__global__ __launch_bounds__(32)
void bsm_scores_kernel(const float* __restrict__ kmat,
                       float* __restrict__ node_max,
                       int*   __restrict__ node_idx)
{
    const int lane = threadIdx.x;
    const int m    = lane & 15;   // row/col within tile
    const int hi   = lane >> 4;   // half-wave selects K pair (2*hi, 2*hi+1)
    const int b    = blockIdx.y;
    const int s0   = blockIdx.x * 16;

    const float* kb   = kmat + (size_t)b * BSM_T * BSM_CK;
    const float* arow = kb + (size_t)(2 * (s0 + m)) * BSM_CK;   // a[s] = k[2s]

    // A tile 16x64 as 16 slices of 16x4; per ISA layout lane holds K=4j+2hi,+1
    v2f A[16];
#pragma unroll
    for (int j = 0; j < 16; ++j)
        A[j] = *(const v2f*)(arow + 4 * j + 2 * hi);

    float lmax[8]; int lidx[8];
#pragma unroll
    for (int i = 0; i < 8; ++i) { lmax[i] = -3.0e38f; lidx[i] = 0; }

#pragma unroll 2
    for (int d0 = 0; d0 < BSM_HALF; d0 += 16) {
        // B tile (b^T): B[k][n] = k[2*(d0+n)+1][k] -> same lane pattern as A.
        // Load whole tile first -> one 16-wide load clause, partial waits.
        const float* brow = kb + (size_t)(2 * (d0 + m) + 1) * BSM_CK;
        v2f Bv[16];
#pragma unroll
        for (int j = 0; j < 16; ++j)
            Bv[j] = *(const v2f*)(brow + 4 * j + 2 * hi);

        v8f c = {0.f, 0.f, 0.f, 0.f, 0.f, 0.f, 0.f, 0.f};
#pragma unroll
        for (int j = 0; j < 16; ++j)
            c = __builtin_amdgcn_wmma_f32_16x16x4_f32(
                    false, A[j], false, Bv[j], (short)0, c, false, false);

        // C layout: lanes 0-15 -> rows s0+i, lanes 16-31 -> rows s0+8+i;
        // column held by this lane = d0 + m. Per-lane running update only
        // (branchless; strict > keeps first occurrence since col increases).
        const int col = d0 + m;
#pragma unroll
        for (int i = 0; i < 8; ++i) {
            const bool gt = c[i] > lmax[i];
            lidx[i] = gt ? col  : lidx[i];
            lmax[i] = gt ? c[i] : lmax[i];
        }
    }

    // Single cross-lane max/argmax reduction over each 16-lane half
    // (masks 1..8 never cross the hi boundary). Tie -> smaller column index,
    // matching jnp.argmax first-occurrence semantics.
#pragma unroll
    for (int i = 0; i < 8; ++i) {
        float v = lmax[i];
        int idx = lidx[i];
#pragma unroll
        for (int msk = 1; msk <= 8; msk <<= 1) {
            const float ov = __shfl_xor(v, msk, 32);
            const int   oi = __shfl_xor(idx, msk, 32);
            const bool take = (ov > v) || (ov == v && oi < idx);
            v   = take ? ov : v;
            idx = take ? oi : idx;
        }
        lmax[i] = v;
        lidx[i] = idx;
    }

    if (m == 0) {  // lane 0 -> rows s0..s0+7, lane 16 -> rows s0+8..s0+15
#pragma unroll
        for (int i = 0; i < 8; ++i) {
            int s = s0 + hi * 8 + i;
            float mv = lmax[i]; int mi = lidx[i];
            if (s == 0) { mv = NEG_INF_F; mi = 0; }  // scores[:,0,:] = -inf
            node_max[(size_t)b * BSM_HALF + s] = mv;
            node_idx[(size_t)b * BSM_HALF + s] = mi;
        }
    }
}

// ---------------------------------------------------------------------------
// Kernel 2: per-batch descending argsort (bitonic, LDS, 64-bit composite keys
// -> exact stable argsort(-node_max)) + CSR of merge lists grouped by dst.
// ---------------------------------------------------------------------------
__global__ __launch_bounds__(256)
void bsm_sort_kernel(const float* __restrict__ node_max,
                     const int*   __restrict__ node_idx,
                     int* __restrict__ edge_idx,     // B x 512
                     int* __restrict__ dst_offsets,  // B x 513
                     int* __restrict__ dst_items)    // B x 256
{
    __shared__ unsigned long long key[BSM_HALF];
    __shared__ int s_dst[BSM_R];
    __shared__ int s_src[BSM_R];
    __shared__ int s_cnt[BSM_HALF];

    const int b = blockIdx.x;
    const int t = threadIdx.x;

    for (int i = t; i < BSM_HALF; i += 256) {
        unsigned int u = __float_as_uint(node_max[(size_t)b * BSM_HALF + i]);
        u = (u & 0x80000000u) ? ~u : (u | 0x80000000u);  // order-preserving map
        u = ~u;                                          // descending
        key[i] = ((unsigned long long)u << 32) | (unsigned int)i;
        s_cnt[i] = 0;
    }
    __syncthreads();

    for (int ksz = 2; ksz <= BSM_HALF; ksz <<= 1) {
        for (int j = ksz >> 1; j > 0; j >>= 1) {
            for (int i = t; i < BSM_HALF; i += 256) {
                int l = i ^ j;
                if (l > i) {
                    bool up = ((i & ksz) == 0);
                    unsigned long long ka = key[i], kc = key[l];
                    if ((ka > kc) == up) { key[i] = kc; key[l] = ka; }
                }
            }
            __syncthreads();
        }
    }

    for (int p = t; p < BSM_HALF; p += 256) {
        int e = (int)(key[p] & 0xffffffffu);
        edge_idx[(size_t)b * BSM_HALF + p] = e;
        if (p < BSM_R) {
            s_src[p] = e;
            s_dst[p] = node_idx[(size_t)b * BSM_HALF + e];
        }
    }
    __syncthreads();

    if (t == 0) {  // serial CSR build: ~1K iterations, negligible
        for (int p = 0; p < BSM_R; ++p) s_cnt[s_dst[p]]++;
        int run = 0;
        for (int d = 0; d < BSM_HALF; ++d) {
            dst_offsets[(size_t)b * (BSM_HALF + 1) + d] = run;
            int c = s_cnt[d];
            s_cnt[d] = run;  // becomes write cursor
            run += c;
        }
        dst_offsets[(size_t)b * (BSM_HALF + 1) + BSM_HALF] = run;
        for (int p = 0; p < BSM_R; ++p) {  // rank order preserved within dst
            int d = s_dst[p];
            dst_items[(size_t)b * BSM_R + (s_cnt[d]++)] = s_src[p];
        }
    }
}

// ---------------------------------------------------------------------------
// Kernel 3: pure-bandwidth output build. One block per output row, 128-bit IO.
// All bulk x reads / out writes are single-use across the whole dispatch, so
// use non-temporal policy (TH=NT) to keep them out of the 192MB L2; only the
// small shared index/CSR reads stay on the temporal path.
// rows [0,256): unm gather. rows [256,768): dst row + ordered merge adds.
// ---------------------------------------------------------------------------
__global__ __launch_bounds__(192)
void bsm_out_kernel(const float* __restrict__ x,
                    const int* __restrict__ edge_idx,
                    const int* __restrict__ dst_offsets,
                    const int* __restrict__ dst_items,
                    float* __restrict__ out)
{
    const int b   = blockIdx.y;
    const int row = blockIdx.x;       // 0..767
    const int t   = threadIdx.x;      // 0..191 (192 * v4f = 768 floats)
    const float* xb = x + (size_t)b * BSM_T * BSM_C;
    v4f* orow = (v4f*)(out + ((size_t)b * 768 + row) * BSM_C);

    if (row < BSM_R) {
        int e = edge_idx[(size_t)b * BSM_HALF + BSM_R + row];
        const v4f* src = (const v4f*)(xb + (size_t)(2 * e) * BSM_C);
        __builtin_nontemporal_store(__builtin_nontemporal_load(src + t),
                                    orow + t);
    } else {
        int d = row - BSM_R;
        const v4f* base = (const v4f*)(xb + (size_t)(2 * d + 1) * BSM_C);
        v4f acc = __builtin_nontemporal_load(base + t);
        int beg = dst_offsets[(size_t)b * (BSM_HALF + 1) + d];
        int end = dst_offsets[(size_t)b * (BSM_HALF + 1) + d + 1];
        for (int q = beg; q < end; ++q) {
            int s = dst_items[(size_t)b * BSM_R + q];
            const v4f* srow = (const v4f*)(xb + (size_t)(2 * s) * BSM_C);
            acc += __builtin_nontemporal_load(srow + t);
        }
        __builtin_nontemporal_store(acc, orow + t);
    }
}

// ---------------------------------------------------------------------------
extern "C" void kernel_launch(void* const* d_in, const int* in_sizes, int n_in,
                              void* d_out, int out_size, void* d_ws, size_t ws_size,
                              hipStream_t stream)
{
    const float* x = (const float*)d_in[0];   // (64, 1024, 768)
    const float* k = (const float*)d_in[1];   // (64, 1024, 64)
    // d_in[2] is r == 256 (compile-time constant BSM_R)

    char* ws = (char*)d_ws;
    float* node_max    = (float*)(ws + 0);        // 64*512*4 = 131072
    int*   node_idx    = (int*)  (ws + 131072);   // 131072
    int*   edge_idx    = (int*)  (ws + 262144);   // 131072
    int*   dst_offsets = (int*)  (ws + 393216);   // 64*513*4 = 131328
    int*   dst_items   = (int*)  (ws + 525312);   // 64*256*4 = 65536
    float* out = (float*)d_out;                   // (64, 768, 768)

    bsm_scores_kernel<<<dim3(BSM_HALF / 16, BSM_B), 32, 0, stream>>>(
        k, node_max, node_idx);
    bsm_sort_kernel<<<dim3(BSM_B), 256, 0, stream>>>(
        node_max, node_idx, edge_idx, dst_offsets, dst_items);
    bsm_out_kernel<<<dim3(768, BSM_B), 192, 0, stream>>>(
        x, edge_idx, dst_offsets, dst_items, out);
}